// HydroNetwork_27444841022129
// MI455X (gfx1250) — compile-verified
//
#include <hip/hip_runtime.h>

// CDNA5 / gfx1250: wave32, WMMA. FP32 matrix path (V_WMMA_F32_16X16X4_F32)
// for 4th-order-derivative numerical sanity.

typedef __attribute__((ext_vector_type(2))) float v2f;
typedef __attribute__((ext_vector_type(8))) float v8f;

#define TILE 16   // samples per workgroup (= WMMA M)
#define HH   128  // hidden width
#define NC   18   // jet components: 1;x,y,t;x2,xy,y2,xt,yt;x3,x2y,xy2,y3,x2t,y2t;x4,x2y2,y4
#define JS   17   // padded LDS sample stride (bank-conflict avoidance)

// Compose tanh with an 18-coefficient truncated Taylor jet (in place).
// J holds Taylor *coefficients* (deriv / multinomial factorial).
__device__ __forceinline__ void tanh_jet(float* J) {
  const float u0 = tanhf(J[0]);
  const float d1 = 1.f - u0 * u0;          // tanh'
  const float d2 = -2.f * u0 * d1;         // tanh''
  const float d3 = -2.f * (d1 * d1 + u0 * d2);
  const float d4 = -2.f * (3.f * d1 * d2 + u0 * d3);
  const float c1 = d1, c2 = 0.5f * d2, c3 = d3 * (1.f / 6.f), c4 = d4 * (1.f / 24.f);

  const float px = J[1], py = J[2], pt = J[3];
  const float pxx = J[4], pxy = J[5], pyy = J[6], pxt = J[7], pyt = J[8];
  const float pxxx = J[9], pxxy = J[10], pxyy = J[11], pyyy = J[12];
  const float pxxt = J[13], pyyt = J[14];
  const float px4 = J[15], pxxyy = J[16], py4 = J[17];
  const float px2 = px * px, py2 = py * py;

  J[0]  = u0;
  J[1]  = c1 * px;
  J[2]  = c1 * py;
  J[3]  = c1 * pt;
  J[4]  = c1 * pxx + c2 * px2;
  J[5]  = c1 * pxy + 2.f * c2 * px * py;
  J[6]  = c1 * pyy + c2 * py2;
  J[7]  = c1 * pxt + 2.f * c2 * px * pt;
  J[8]  = c1 * pyt + 2.f * c2 * py * pt;
  J[9]  = c1 * pxxx + 2.f * c2 * px * pxx + c3 * px2 * px;
  J[10] = c1 * pxxy + 2.f * c2 * (px * pxy + py * pxx) + 3.f * c3 * px2 * py;
  J[11] = c1 * pxyy + 2.f * c2 * (py * pxy + px * pyy) + 3.f * c3 * px * py2;
  J[12] = c1 * pyyy + 2.f * c2 * py * pyy + c3 * py2 * py;
  J[13] = c1 * pxxt + 2.f * c2 * (px * pxt + pt * pxx) + 3.f * c3 * px2 * pt;
  J[14] = c1 * pyyt + 2.f * c2 * (py * pyt + pt * pyy) + 3.f * c3 * py2 * pt;
  J[15] = c1 * px4 + c2 * (2.f * px * pxxx + pxx * pxx) + 3.f * c3 * px2 * pxx + c4 * px2 * px2;
  J[16] = c1 * pxxyy
        + c2 * (2.f * px * pxyy + 2.f * py * pxxy + 2.f * pxx * pyy + pxy * pxy)
        + c3 * (3.f * px2 * pyy + 6.f * px * py * pxy + 3.f * py2 * pxx)
        + 6.f * c4 * px2 * py2;
  J[17] = c1 * py4 + c2 * (2.f * py * pyyy + pyy * pyy) + 3.f * c3 * py2 * pyy + c4 * py2 * py2;
}

extern __shared__ float smem[];

__global__ __launch_bounds__(256)
void hydro_jet_kernel(const float* __restrict__ x,
                      const float* __restrict__ W1, const float* __restrict__ b1,
                      const float* __restrict__ W2, const float* __restrict__ b2,
                      const float* __restrict__ W3, const float* __restrict__ b3,
                      const float* __restrict__ W4,
                      const float* __restrict__ nu,
                      float* __restrict__ out, int n) {
  float* jet = smem;                 // [NC][HH][JS] fp32 jet state, in-place per layer
  float* w4s = jet + NC * HH * JS;   // [HH]
  float* Fm  = w4s + HH;             // [NC][TILE] final output jet
  float* xs  = Fm + NC * TILE;       // [TILE][3]

  const int t    = threadIdx.x;
  const int lane = t & 31;
  const int wv   = t >> 5;          // wave id: 8 waves x 16 output columns
  const int sl   = lane & 15;
  const int hi   = lane >> 4;       // half-wave select
  const int koff = hi << 1;         // A/B K offset per documented 16x4 layout
  const int roff = hi << 3;         // C/D row offset per documented layout
  const int nb   = wv << 4;
  const int s0   = blockIdx.x * TILE;

  if (t < 3 * TILE) xs[t] = x[s0 * 3 + t];
  if (t < HH)       w4s[t] = W4[t];
  __syncthreads();

  // ---- layer 1: linear seed (exact jet) + tanh ----
  for (int p = t; p < HH * TILE; p += 256) {
    const int j = p >> 4, s = p & 15;
    const float px = W1[j], py = W1[HH + j], pt = W1[2 * HH + j];
    float J[NC];
    J[0] = xs[3 * s] * px + xs[3 * s + 1] * py + xs[3 * s + 2] * pt + b1[j];
    J[1] = px; J[2] = py; J[3] = pt;
#pragma unroll
    for (int c = 4; c < NC; ++c) J[c] = 0.f;
    tanh_jet(J);
#pragma unroll
    for (int c = 0; c < NC; ++c) jet[(c * HH + j) * JS + s] = J[c];
  }
  __syncthreads();

  // ---- hidden layers 2 & 3: per-component 16x128x128 GEMM via WMMA f32 ----
  const float* Wg_[2] = {W2, W3};
  const float* bg_[2] = {b2, b3};
  for (int L = 0; L < 2; ++L) {
    const float* __restrict__ Wg = Wg_[L];
    // Wave-resident B fragments: this wave's 16 columns of W, all 32 K-steps.
    v2f bfrag[32];
#pragma unroll
    for (int kk = 0; kk < 32; ++kk) {
      const int row = 4 * kk + koff;
      bfrag[kk].x = Wg[row * HH + nb + sl];
      bfrag[kk].y = Wg[(row + 1) * HH + nb + sl];
    }
    const float bconst = bg_[L][nb + sl];

    for (int c = 0; c < NC; ++c) {
      v8f acc = {};
#pragma unroll
      for (int kk = 0; kk < 32; ++kk) {
        const int k0 = 4 * kk + koff;
        v2f a;
        a.x = jet[(c * HH + k0) * JS + sl];
        a.y = jet[(c * HH + k0 + 1) * JS + sl];
        acc = __builtin_amdgcn_wmma_f32_16x16x4_f32(
            /*neg_a=*/false, a, /*neg_b=*/false, bfrag[kk],
            /*c_mod=*/(short)0, acc, /*reuse_a=*/false, /*reuse_b=*/false);
      }
      __syncthreads();  // all waves finished reading component c (WAR)
      const float bv = (c == 0) ? bconst : 0.f;  // bias only hits constant comp
#pragma unroll
      for (int r = 0; r < 8; ++r)
        jet[(c * HH + nb + sl) * JS + (r + roff)] = acc[r] + bv;
      // next component reads disjoint LDS; no barrier needed here
    }
    __syncthreads();

    // tanh-jet recomposition, element-wise over (sample, neuron)
    for (int p = t; p < HH * TILE; p += 256) {
      const int j = p >> 4, s = p & 15;
      float J[NC];
#pragma unroll
      for (int c = 0; c < NC; ++c) J[c] = jet[(c * HH + j) * JS + s];
      tanh_jet(J);
#pragma unroll
      for (int c = 0; c < NC; ++c) jet[(c * HH + j) * JS + s] = J[c];
    }
    __syncthreads();
  }

  // ---- layer 4: 128 -> 1 reduction per (component, sample) ----
  {
    const int c = t >> 4, s = t & 15;  // comps 0..15
    float acc = 0.f;
    for (int j = 0; j < HH; ++j) acc += jet[(c * HH + j) * JS + s] * w4s[j];
    Fm[c * TILE + s] = acc;
  }
  if (t < 32) {
    const int c = 16 + (t >> 4), s = t & 15;  // comps 16,17
    float acc = 0.f;
    for (int j = 0; j < HH; ++j) acc += jet[(c * HH + j) * JS + s] * w4s[j];
    Fm[c * TILE + s] = acc;
  }
  __syncthreads();

  // ---- derivative extraction + NSE residual (b4 cancels in all derivatives) ----
  if (t < TILE) {
    const int s = t, gi = s0 + s;
    const float psi_x   = Fm[1 * TILE + s];
    const float psi_y   = Fm[2 * TILE + s];
    const float psi_xxx = 6.f  * Fm[9 * TILE + s];
    const float psi_xxy = 2.f  * Fm[10 * TILE + s];
    const float psi_xyy = 2.f  * Fm[11 * TILE + s];
    const float psi_yyy = 6.f  * Fm[12 * TILE + s];
    const float psi_xxt = 2.f  * Fm[13 * TILE + s];
    const float psi_yyt = 2.f  * Fm[14 * TILE + s];
    const float psi_x4  = 24.f * Fm[15 * TILE + s];
    const float psi_xxyy = 4.f * Fm[16 * TILE + s];
    const float psi_y4  = 24.f * Fm[17 * TILE + s];

    const float u = psi_y;
    const float v = -psi_x;
    const float wx = -(psi_xxx + psi_xyy);
    const float wy = -(psi_xxy + psi_yyy);
    const float wt = -(psi_xxt + psi_yyt);
    const float lapw = -(psi_x4 + 2.f * psi_xxyy + psi_y4);
    const float nse = wt + wx * u + wy * v - nu[0] * lapw;

    out[2 * gi]     = u;
    out[2 * gi + 1] = v;
    out[2 * n + gi] = nse;
  }
}

extern "C" void kernel_launch(void* const* d_in, const int* in_sizes, int n_in,
                              void* d_out, int out_size, void* d_ws, size_t ws_size,
                              hipStream_t stream) {
  (void)n_in; (void)out_size; (void)d_ws; (void)ws_size;
  const float* x  = (const float*)d_in[0];
  const float* W1 = (const float*)d_in[1];
  const float* b1 = (const float*)d_in[2];
  const float* W2 = (const float*)d_in[3];
  const float* b2 = (const float*)d_in[4];
  const float* W3 = (const float*)d_in[5];
  const float* b3 = (const float*)d_in[6];
  const float* W4 = (const float*)d_in[7];
  // d_in[8] = b4: contributes only to psi's value, not to any output derivative
  const float* nu = (const float*)d_in[9];
  float* out = (float*)d_out;

  const int n = in_sizes[0] / 3;  // N = 8192
  const size_t shmem =
      (size_t)(NC * HH * JS + HH + NC * TILE + 3 * TILE) * sizeof(float);  // ~155 KB
  (void)hipFuncSetAttribute(reinterpret_cast<const void*>(&hydro_jet_kernel),
                            hipFuncAttributeMaxDynamicSharedMemorySize,
                            (int)shmem);
  hydro_jet_kernel<<<n / TILE, 256, shmem, stream>>>(
      x, W1, b1, W2, b2, W3, b3, W4, nu, out, n);
}